// LinearPredictiveCoefficientsToLineSpectralPairs_30399778521122
// MI455X (gfx1250) — compile-verified
//
#include <hip/hip_runtime.h>
#include <math.h>

// LPC -> LSP for 128000 independent order-16 problems.
// Stage 1: p,q coefficient construction as 16x16x16 f32 GEMM via
//          V_WMMA_F32_16X16X4_F32 (16 problems per wave, wave 0 of block).
// Stage 2: Durand-Kerner simultaneous root iteration, 1 problem per wave,
//          lanes 0-15 = roots of p, lanes 16-31 = roots of q, cross-lane
//          shuffles (ds_bpermute) for the product terms. Angle selection +
//          rank sort reproduces "every other eigenvalue" + jnp.sort.

typedef __attribute__((ext_vector_type(2))) float v2f;
typedef __attribute__((ext_vector_type(8))) float v8f;

#define NC 17               // coefficients per polynomial (degree 16)
#define PROB_PER_BLOCK 16
#define THREADS 512         // 16 waves of 32
#define DK_ITERS 64

// Mp[j][t] = [t<=j] - [t>=17-j]   (p = Mp . a0, descending-power coeffs)
__device__ __forceinline__ float mp_elem(int j, int t) {
    float v = 0.0f;
    if (t <= j)      v += 1.0f;
    if (t >= 17 - j) v -= 1.0f;
    return v;
}
// Mq[j][t] = (-1)^(j-t) * Mp[j][t]
__device__ __forceinline__ float mq_elem(int j, int t) {
    float v = mp_elem(j, t);
    return ((j - t) & 1) ? -v : v;
}

__global__ __launch_bounds__(THREADS)
void lpc_to_lsp_kernel(const float* __restrict__ a, float* __restrict__ out, int nprob)
{
    __shared__ float sP[PROB_PER_BLOCK][NC];   // p coeffs per problem
    __shared__ float sQ[PROB_PER_BLOCK][NC];   // q coeffs per problem
    __shared__ float sSel[PROB_PER_BLOCK][16]; // selected angles (8 from p, 8 from q)

    const int tid   = threadIdx.x;
    const int pbase = blockIdx.x * PROB_PER_BLOCK;

    // ---------------- Stage 1: WMMA construction of p and q (wave 0) -------
    if (tid < 32) {
        const int lane = tid;
        const int nidx = lane & 15;   // problem column (B/D) and matrix row (A)
        const int hi   = lane >> 4;   // which K-pair this lane supplies
        const int pidx = pbase + nidx;
        const int pc   = (pidx < nprob) ? pidx : (nprob - 1);
        const float* ap = a + (long)pc * NC;

        __builtin_prefetch(ap, 0, 1);   // global_prefetch_b8

        const float a16 = ap[16];

        v8f accP = {}; v8f accQ = {};
        // D(16x16) = M(16x16) * A0(16x16), K=16 as 4 chained K=4 WMMAs.
        // A layout: lanes 0-15 row M, VGPR0=K(k0+2*hi), VGPR1=K+1 (ISA 7.12.2).
        // B layout mirrors: lane supplies column n=nidx, same K pattern.
        #pragma unroll
        for (int k0 = 0; k0 < 16; k0 += 4) {
            const int kA = k0 + 2 * hi;
            v2f bf, aPf, aQf;
            bf.x  = (kA == 0) ? 1.0f : ap[kA];      // a0 = [1, a1..a15]
            bf.y  = ap[kA + 1];
            aPf.x = mp_elem(nidx, kA); aPf.y = mp_elem(nidx, kA + 1);
            aQf.x = mq_elem(nidx, kA); aQf.y = mq_elem(nidx, kA + 1);
            accP = __builtin_amdgcn_wmma_f32_16x16x4_f32(
                false, aPf, false, bf, (short)0, accP, false, false);
            accQ = __builtin_amdgcn_wmma_f32_16x16x4_f32(
                false, aQf, false, bf, (short)0, accQ, false, false);
        }

        // rank-1 fixup for the 17th coefficient a0[16]:
        //   p[j] += Mp[j][16]*a16 = -a16          (j>=1)
        //   q[j] += Mq[j][16]*a16 = -(-1)^j a16   (j>=1)
        // D layout: lane holds D[v + 8*hi][nidx] in accX[v].
        #pragma unroll
        for (int v = 0; v < 8; v++) {
            const int j = v + 8 * hi;
            float pv = accP[v];
            float qv = accQ[v];
            if (j >= 1) {
                pv -= a16;
                qv += (j & 1) ? a16 : -a16;
            }
            sP[nidx][j] = pv;
            sQ[nidx][j] = qv;
        }
        if (hi == 0) {
            sP[nidx][16] = 1.0f;   // palindromic: p[16] = p[0] = 1 exactly
            sQ[nidx][16] = 1.0f;
            if (pidx < nprob) out[(long)pidx * NC] = ap[0];   // K passthrough
        }
    }
    __syncthreads();

    // ---------------- Stage 2: Durand-Kerner, one problem per wave ---------
    const int wave  = tid >> 5;
    const int lane  = tid & 31;
    const int half  = lane >> 4;        // 0: poly p, 1: poly q
    const int k     = lane & 15;        // root index within polynomial
    const int base  = half << 4;        // shuffle segment base
    const int pidx2 = pbase + wave;
    const bool valid = (pidx2 < nprob);

    float c[NC];
    #pragma unroll
    for (int j = 0; j < NC; j++)
        c[j] = half ? sQ[wave][j] : sP[wave][j];

    // classic DK init: z_k = (0.4 + 0.9i)^(k+1)
    float zr = 1.0f, zi = 0.0f;
    for (int i = 0; i <= k; i++) {
        const float t = zr * 0.4f - zi * 0.9f;
        zi = zr * 0.9f + zi * 0.4f;
        zr = t;
    }

    for (int it = 0; it < DK_ITERS; it++) {
        // Horner, descending powers (companion-matrix convention)
        float pr = c[0], pi = 0.0f;
        #pragma unroll
        for (int j = 1; j < NC; j++) {
            const float t = pr * zr - pi * zi + c[j];
            pi = pr * zi + pi * zr;
            pr = t;
        }
        // denom = c0 * prod_{j!=k} (z_k - z_j), z_j via cross-lane shuffle
        float dr = c[0], di = 0.0f;
        #pragma unroll
        for (int j = 0; j < 16; j++) {
            const float orr = __shfl(zr, base + j, 32);
            const float oii = __shfl(zi, base + j, 32);
            if (j != k) {
                const float exr = zr - orr, exi = zi - oii;
                const float t = dr * exr - di * exi;
                di = dr * exi + di * exr;
                dr = t;
            }
        }
        const float den = dr * dr + di * di + 1e-30f;
        zr -= (pr * dr + pi * di) / den;
        zi -= (pi * dr - pr * di) / den;
    }

    // one angle per conjugate pair: |atan2|, then keep even ranks (dedup)
    const float ang = fabsf(atan2f(zi, zr));
    int rank = 0;
    #pragma unroll
    for (int j = 0; j < 16; j++) {
        const float aj = __shfl(ang, base + j, 32);
        rank += (aj < ang) || (aj == ang && j < k);
    }
    if ((rank & 1) == 0)
        sSel[wave][(half << 3) + (rank >> 1)] = ang;
    __syncthreads();

    if (((rank & 1) == 0) && valid) {
        const int myt = (half << 3) + (rank >> 1);
        int fin = 0;
        #pragma unroll
        for (int t = 0; t < 16; t++) {
            const float v = sSel[wave][t];
            fin += (v < ang) || (v == ang && t < myt);
        }
        out[(long)pidx2 * NC + 1 + fin] = ang;   // positions 1..16, sorted
    }
}

extern "C" void kernel_launch(void* const* d_in, const int* in_sizes, int n_in,
                              void* d_out, int out_size, void* d_ws, size_t ws_size,
                              hipStream_t stream) {
    const float* a  = (const float*)d_in[0];
    float*       o  = (float*)d_out;
    const int nprob = in_sizes[0] / NC;            // 64*2000 = 128000
    const int blocks = (nprob + PROB_PER_BLOCK - 1) / PROB_PER_BLOCK;
    lpc_to_lsp_kernel<<<blocks, THREADS, 0, stream>>>(a, o, nprob);
}